// EnvelopeAR_34909494182163
// MI455X (gfx1250) — compile-verified
//
#include <hip/hip_runtime.h>
#include <cstdint>
#include <cstddef>

// ---------------------------------------------------------------------------
// EnvelopeAR: e_t = x_t + alpha_t*(e_{t-1}-x_t),
//   alpha_t = s*alpha_a + (1-s)*alpha_r,  s = sigmoid(K*(x_t - e_{t-1}))
// Nonlinear recurrence -> strictly serial in t; parallel only across B=256
// channels. One lane per channel; async global->LDS double-buffered staging.
// Recurrent critical cycle per step (gfx1250): v_fma -> v_tanh_f32 -> v_fma.
// ---------------------------------------------------------------------------

#define ENV_FS    48000.0f
#define ENV_K     50.0f
#define ENV_TPB   32
#define ENV_CHUNK 64   // samples staged per chunk (32 async b64 ops per chunk)
#define ENV_PAD   4    // dword pad -> 272B rows: 16B-aligned for ds_load_b128

// -------- CDNA5 async global->LDS staging (guarded) ------------------------
#if defined(__gfx1250__) && __has_builtin(__builtin_amdgcn_global_load_async_to_lds_b64)
#define ENV_USE_ASYNC 1
#endif

// builtin signature (from clang diagnostic): v2i32 AS1* src, v2i32 AS3* dst
typedef int env_v2i __attribute__((vector_size(2 * sizeof(int))));
typedef __attribute__((address_space(1))) env_v2i* env_g64p;
typedef __attribute__((address_space(3))) env_v2i* env_l64p;

// native clang vector for b128 LDS preloads and NT b128 stores
typedef float env_v4f __attribute__((ext_vector_type(4)));

__device__ __forceinline__ void env_stage_chunk(const float* __restrict__ g,
                                                float* __restrict__ l) {
#if defined(ENV_USE_ASYNC)
#pragma unroll
  for (int k = 0; k < ENV_CHUNK; k += 2) {
    __builtin_amdgcn_global_load_async_to_lds_b64(
        (env_g64p)(g + k),
        (env_l64p)(l + k),
        /*offset=*/0, /*cpol=*/0);
  }
#else
#pragma unroll
  for (int k = 0; k < ENV_CHUNK; k += 2) {
    l[k]     = g[k];
    l[k + 1] = g[k + 1];
  }
#endif
}

__device__ __forceinline__ void env_wait_async_pending() {
#if defined(ENV_USE_ASYNC)
#if __has_builtin(__builtin_amdgcn_s_wait_asynccnt)
  __builtin_amdgcn_s_wait_asynccnt(ENV_CHUNK / 2);  // allow next chunk in flight
#else
  asm volatile("s_wait_asynccnt 32" ::: "memory");
#endif
  asm volatile("" ::: "memory");  // keep LDS reads after the wait
#endif
}

__device__ __forceinline__ void env_wait_async_zero() {
#if defined(ENV_USE_ASYNC)
#if __has_builtin(__builtin_amdgcn_s_wait_asynccnt)
  __builtin_amdgcn_s_wait_asynccnt(0);
#else
  asm volatile("s_wait_asynccnt 0" ::: "memory");
#endif
  asm volatile("" ::: "memory");
#endif
}

// -------- fast transcendental helpers --------------------------------------
#if defined(__gfx1250__) && __has_builtin(__builtin_amdgcn_tanhf)
#define ENV_HAVE_TANH 1
#endif

__device__ __forceinline__ float env_tanh(float v) {
#if defined(ENV_HAVE_TANH)
  return __builtin_amdgcn_tanhf(v);   // v_tanh_f32 (new on gfx1250)
#else
  return tanhf(v);
#endif
}

__device__ __forceinline__ float env_exp2(float v) {
#if defined(__gfx1250__) && __has_builtin(__builtin_amdgcn_exp2f)
  return __builtin_amdgcn_exp2f(v);   // v_exp_f32
#else
  return exp2f(v);
#endif
}

__device__ __forceinline__ float env_rcp(float v) {
#if defined(__gfx1250__) && __has_builtin(__builtin_amdgcn_rcpf)
  return __builtin_amdgcn_rcpf(v);    // v_rcp_f32
#else
  return 1.0f / v;
#endif
}

// ---------------------------------------------------------------------------
// Step math.  m = e - x; s = sigmoid(-K*m) = 0.5 - 0.5*tanh((K/2)*m).
//   e' = x + (ar + da*s)*m
//      = [x + 0.5*(aa+ar)*m] + [-0.5*da*m]*tanh((K/2)*m)
// Distribute over e so all x-only terms precompute off the recurrent cycle:
//   t0 = C*e + c0        c0 = -C*x          (C = K/2)
//   u  = Amid*e + u0     u0 = x - Amid*x    (Amid = 0.5*(aa+ar))
//   w  = nhda*e + w0     w0 = -nhda*x       (nhda = -0.5*(aa-ar))
//   e' = tanh(t0)*w + u
// Recurrent cycle: fma -> tanh -> fma (u,w overlap the tanh latency).
// ---------------------------------------------------------------------------
#define ENV_C (0.5f * ENV_K)

__device__ __forceinline__ float env_step_pre(float e, float c0, float u0, float w0,
                                              float Amid, float nhda) {
#if defined(ENV_HAVE_TANH)
  const float t0 = fmaf(ENV_C, e, c0);   // cycle op 1
  const float u  = fmaf(Amid, e, u0);    // || with tanh
  const float w  = fmaf(nhda, e, w0);    // || with tanh
  const float th = env_tanh(t0);         // cycle op 2
  return fmaf(th, w, u);                 // cycle op 3
#else
  // fallback: tanh(z) = 1 - 2/(1 + exp2(2*log2e*z)), z = t0 = C*(e-x)
  const float t0 = fmaf(ENV_C, e, c0);
  const float p  = env_exp2((2.0f * 1.4426950408889634f) * t0);
  const float s  = env_rcp(1.0f + p);
  const float u  = fmaf(Amid, e, u0);
  const float w  = fmaf(nhda, e, w0);
  const float th = fmaf(-2.0f, s, 1.0f);
  return fmaf(th, w, u);
#endif
}

// ---------------------------------------------------------------------------
__global__ __launch_bounds__(ENV_TPB)
void EnvelopeAR_34909494182163_kernel(const float* __restrict__ x_env,
                                      const float* __restrict__ tau_a,
                                      const float* __restrict__ tau_r,
                                      float* __restrict__ out,
                                      int B, int T) {
  // +4 dword pad: rows are 272B (16B-aligned -> ds_load_b128 preloads, and
  // 8B-aligned for async b64 staging). Banks (4*lane+t)%64: <=2-way conflict.
  __shared__ float buf[2][ENV_TPB][ENV_CHUNK + ENV_PAD];

  const int tid = threadIdx.x;
  const int b   = blockIdx.x * ENV_TPB + tid;
  if (b >= B) return;

  // per-channel constants: alpha = exp(-1/(tau*fs)) = exp2(-log2e/(tau*fs))
  const float aa   = env_exp2(-1.4426950408889634f / (tau_a[b] * ENV_FS));
  const float ar   = env_exp2(-1.4426950408889634f / (tau_r[b] * ENV_FS));
  const float da   = aa - ar;
  const float Amid = 0.5f * (aa + ar);   // ar + 0.5*da
  const float nhda = -0.5f * da;

  const float* __restrict__ xrow = x_env + (size_t)b * (size_t)T;
  float* __restrict__       orow = out   + (size_t)b * (size_t)T;

  float e = 0.0f;
  const int nchunks = T / ENV_CHUNK;

  if (nchunks > 0) {
    env_stage_chunk(xrow, &buf[0][tid][0]);  // prologue: stage chunk 0
  }

  for (int c = 0; c < nchunks; ++c) {
    const int cur = c & 1;
    if (c + 1 < nchunks) {
      env_stage_chunk(xrow + (size_t)(c + 1) * ENV_CHUNK, &buf[cur ^ 1][tid][0]);
      env_wait_async_pending();   // chunk c resident; chunk c+1 still in flight
    } else {
      env_wait_async_zero();
    }

    const float* __restrict__ lrow = &buf[cur][tid][0];
    float* __restrict__ op = orow + (size_t)c * ENV_CHUNK;

    // Preload the whole chunk: 16 back-to-back ds_load_b128 into VGPRs so the
    // serial tanh chain below never waits on DScnt mid-chunk.
    env_v4f xq[ENV_CHUNK / 4];
#pragma unroll
    for (int g = 0; g < ENV_CHUNK / 4; ++g) {
      xq[g] = reinterpret_cast<const env_v4f*>(lrow)[g];
    }

#pragma unroll
    for (int g = 0; g < ENV_CHUNK / 4; ++g) {
      // preprocess 4 samples off the recurrent cycle
      float c0[4], u0[4], w0[4];
#pragma unroll
      for (int j = 0; j < 4; ++j) {
        const float xv = xq[g][j];
        c0[j] = -ENV_C * xv;
        u0[j] = fmaf(-Amid, xv, xv);
        w0[j] = -nhda * xv;
      }
      env_v4f o4;
#pragma unroll
      for (int j = 0; j < 4; ++j) {
        e = env_step_pre(e, c0[j], u0[j], w0[j], Amid, nhda);
        o4[j] = e;
      }
      // write-once stream: non-temporal b128 store
      __builtin_nontemporal_store(o4, reinterpret_cast<env_v4f*>(op + 4 * g));
    }
  }

  // tail (T not a multiple of ENV_CHUNK): straight global loads
  for (int t = nchunks * ENV_CHUNK; t < T; ++t) {
    const float xv = xrow[t];
    e = env_step_pre(e, -ENV_C * xv, fmaf(-Amid, xv, xv), -nhda * xv, Amid, nhda);
    orow[t] = e;
  }
}

// ---------------------------------------------------------------------------
extern "C" void kernel_launch(void* const* d_in, const int* in_sizes, int n_in,
                              void* d_out, int out_size, void* d_ws, size_t ws_size,
                              hipStream_t stream) {
  (void)n_in; (void)d_ws; (void)ws_size; (void)out_size;

  const float* x_env = (const float*)d_in[0];
  const float* tau_a = (const float*)d_in[1];
  const float* tau_r = (const float*)d_in[2];
  float* out = (float*)d_out;

  const int B = in_sizes[1];                 // tau_a has B elements
  const int T = (B > 0) ? (in_sizes[0] / B) : 0;

  const int blocks = (B + ENV_TPB - 1) / ENV_TPB;
  if (blocks > 0 && T > 0) {
    EnvelopeAR_34909494182163_kernel<<<blocks, ENV_TPB, 0, stream>>>(
        x_env, tau_a, tau_r, out, B, T);
  }
}